// GCN_5LAYERS_58033598103987
// MI455X (gfx1250) — compile-verified
//
#include <hip/hip_runtime.h>
#include <math.h>

typedef __attribute__((ext_vector_type(2))) float v2f;
typedef __attribute__((ext_vector_type(8))) float v8f;

#define NODES 200000
#define THREADS 256

// ---------------------------------------------------------------------------
// float atomic max -> should lower to global_atomic_max_num_f32 on gfx1250
// ---------------------------------------------------------------------------
__device__ __forceinline__ void atomicMaxF(float* addr, float v) {
    (void)__hip_atomic_fetch_max(addr, v, __ATOMIC_RELAXED, __HIP_MEMORY_SCOPE_AGENT);
}

// ---------------------------------------------------------------------------
// degree / normalization
// ---------------------------------------------------------------------------
__global__ void init_ones_kernel(float* __restrict__ deg, int n) {
    int i = blockIdx.x * blockDim.x + threadIdx.x;
    if (i < n) deg[i] = 1.0f;                       // self loop contributes 1
}

__global__ void deg_accum_kernel(const int* __restrict__ col, float* __restrict__ deg, int E) {
    int e = blockIdx.x * blockDim.x + threadIdx.x;
    if (e < E) atomicAdd(&deg[col[e]], 1.0f);
}

__global__ void rsqrt_kernel(float* __restrict__ d, int n) {
    int i = blockIdx.x * blockDim.x + threadIdx.x;
    if (i < n) d[i] = rsqrtf(d[i]);                 // deg >= 1 always (self loops)
}

// ---------------------------------------------------------------------------
// WMMA GEMM:  XL = H @ W   (H: nNodes x D_IN, W: D_IN x D_OUT row-major)
// Also seeds AGG with the self-loop message: AGG[i,k] = dinv[i]^2 * XL[i,k]
// One wave per 16-row tile, f32 WMMA 16x16x4, W staged zero-padded in LDS.
// ---------------------------------------------------------------------------
template <int D_IN, int D_OUT>
__global__ __launch_bounds__(THREADS)
void gemm_wmma_kernel(const float* __restrict__ H, const float* __restrict__ W,
                      const float* __restrict__ dinv, float* __restrict__ XL,
                      float* __restrict__ AGG, int nNodes) {
    constexpr int KP = (D_IN + 3) & ~3;             // K padded to multiple of 4
    constexpr int NT = (D_OUT + 15) / 16;           // 16-wide column tiles
    constexpr int NP = NT * 16;
    __shared__ float Wl[KP][NP];

    // cooperative zero-padded load of W into LDS
    for (int idx = threadIdx.x; idx < KP * NP; idx += blockDim.x) {
        int k = idx / NP, n = idx % NP;
        Wl[k][n] = (k < D_IN && n < D_OUT) ? W[k * D_OUT + n] : 0.0f;
    }
    __syncthreads();

    const int wave = threadIdx.x >> 5;
    const int lane = threadIdx.x & 31;
    const int tileRow = (blockIdx.x * (blockDim.x >> 5) + wave) * 16;
    if (tileRow >= nNodes) return;                  // wave-uniform: EXEC stays full

    const int m     = lane & 15;                    // row (A) / col (B) within tile
    const int khalf = (lane >> 4) << 1;             // 0 or 2
    const float* __restrict__ hrow = H + (size_t)(tileRow + m) * D_IN;

    #pragma unroll
    for (int t = 0; t < NT; ++t) {
        v8f c = {};
        const int n = t * 16 + m;                   // B-fragment column
        #pragma unroll
        for (int kb = 0; kb < KP; kb += 4) {
            const int k0 = kb + khalf;
            v2f a, b;
            a.x = (k0     < D_IN) ? hrow[k0]     : 0.0f;
            a.y = (k0 + 1 < D_IN) ? hrow[k0 + 1] : 0.0f;
            b.x = Wl[k0][n];
            b.y = Wl[k0 + 1][n];
            c = __builtin_amdgcn_wmma_f32_16x16x4_f32(
                    false, a, false, b, (short)0, c, false, false);
        }
        const int col = t * 16 + (lane & 15);
        if (col < D_OUT) {
            const int rbase = tileRow + ((lane >> 4) << 3);
            #pragma unroll
            for (int v = 0; v < 8; ++v) {
                const int r  = rbase + v;
                const float x = c[v];
                XL[(size_t)r * D_OUT + col] = x;
                const float dv = dinv[r];
                AGG[(size_t)r * D_OUT + col] = dv * dv * x;   // self-loop message
            }
        }
    }
}

// ---------------------------------------------------------------------------
// edge scatter-max:  AGG[col] = max(AGG[col], dinv[row]*dinv[col] * XL[row])
// ---------------------------------------------------------------------------
template <int D>
__global__ void edge_max_kernel(const int* __restrict__ row, const int* __restrict__ col,
                                const float* __restrict__ dinv, const float* __restrict__ XL,
                                float* __restrict__ AGG, int E) {
    int e = blockIdx.x * blockDim.x + threadIdx.x;
    if (e >= E) return;
    const int r = row[e], c = col[e];
    const float nrm = dinv[r] * dinv[c];
    const float* __restrict__ xr = XL + (size_t)r * D;
    float* __restrict__ ac = AGG + (size_t)c * D;
    if constexpr ((D & 3) == 0) {
        const float4* x4 = (const float4*)xr;       // row stride D*4 bytes, 16B aligned
        #pragma unroll
        for (int q = 0; q < D / 4; ++q) {
            float4 v = x4[q];
            atomicMaxF(ac + 4 * q + 0, nrm * v.x);
            atomicMaxF(ac + 4 * q + 1, nrm * v.y);
            atomicMaxF(ac + 4 * q + 2, nrm * v.z);
            atomicMaxF(ac + 4 * q + 3, nrm * v.w);
        }
    } else {
        #pragma unroll
        for (int k = 0; k < D; ++k) atomicMaxF(ac + k, nrm * xr[k]);
    }
}

// ---------------------------------------------------------------------------
// h = tanh(agg + b)
// ---------------------------------------------------------------------------
template <int D>
__global__ void tanh_bias_kernel(const float* __restrict__ AGG, const float* __restrict__ b,
                                 float* __restrict__ Hout, int nNodes) {
    int i = blockIdx.x * blockDim.x + threadIdx.x;
    if (i >= nNodes * D) return;
    Hout[i] = tanhf(AGG[i] + b[i % D]);
}

// ---------------------------------------------------------------------------
// out = h @ Wc + bc  (2 -> 4), plus copy of h into the second output slot
// ---------------------------------------------------------------------------
__global__ void classifier_kernel(const float* __restrict__ H, const float* __restrict__ Wc,
                                  const float* __restrict__ bc, float* __restrict__ out,
                                  float* __restrict__ hcopy, int nNodes) {
    int n = blockIdx.x * blockDim.x + threadIdx.x;
    if (n >= nNodes) return;
    const float h0 = H[2 * n], h1 = H[2 * n + 1];
    #pragma unroll
    for (int c = 0; c < 4; ++c)
        out[4 * n + c] = fmaf(h0, Wc[c], fmaf(h1, Wc[4 + c], bc[c]));
    hcopy[2 * n]     = h0;
    hcopy[2 * n + 1] = h1;
}

// ---------------------------------------------------------------------------
// launcher
// ---------------------------------------------------------------------------
template <int D_IN, int D_OUT>
static void run_layer(const float* H, const float* W, const float* b,
                      const int* row, const int* col, const float* dinv,
                      float* bufX, float* bufG, float* bufH,
                      int n, int E, hipStream_t stream) {
    const int tiles = (n + 15) / 16;
    const int gblocks = (tiles + 7) / 8;            // 8 waves per 256-thread block
    gemm_wmma_kernel<D_IN, D_OUT><<<gblocks, THREADS, 0, stream>>>(H, W, dinv, bufX, bufG, n);
    edge_max_kernel<D_OUT><<<(E + THREADS - 1) / THREADS, THREADS, 0, stream>>>(row, col, dinv, bufX, bufG, E);
    tanh_bias_kernel<D_OUT><<<((size_t)n * D_OUT + THREADS - 1) / THREADS, THREADS, 0, stream>>>(bufG, b, bufH, n);
}

extern "C" void kernel_launch(void* const* d_in, const int* in_sizes, int n_in,
                              void* d_out, int out_size, void* d_ws, size_t ws_size,
                              hipStream_t stream) {
    const float* x  = (const float*)d_in[0];
    const int*   ei = (const int*)d_in[1];
    const float* W1 = (const float*)d_in[2];  const float* b1 = (const float*)d_in[3];
    const float* W2 = (const float*)d_in[4];  const float* b2 = (const float*)d_in[5];
    const float* W3 = (const float*)d_in[6];  const float* b3 = (const float*)d_in[7];
    const float* W4 = (const float*)d_in[8];  const float* b4 = (const float*)d_in[9];
    const float* W5 = (const float*)d_in[10]; const float* b5 = (const float*)d_in[11];
    const float* Wc = (const float*)d_in[12]; const float* bc = (const float*)d_in[13];

    const int n = in_sizes[0] / 128;                // 200000
    const int E = in_sizes[1] / 2;                  // 6400000
    const int* row = ei;                            // edge_index[0]
    const int* col = ei + E;                        // edge_index[1]

    float* dinv = (float*)d_ws;
    float* bufH = dinv + n;                         // h       (<= N*24 floats)
    float* bufX = bufH + (size_t)n * 24;            // xl
    float* bufG = bufX + (size_t)n * 24;            // agg

    // symmetric normalization: dinv = rsqrt(deg(A+I))
    init_ones_kernel<<<(n + THREADS - 1) / THREADS, THREADS, 0, stream>>>(dinv, n);
    deg_accum_kernel<<<(E + THREADS - 1) / THREADS, THREADS, 0, stream>>>(col, dinv, E);
    rsqrt_kernel<<<(n + THREADS - 1) / THREADS, THREADS, 0, stream>>>(dinv, n);

    run_layer<128, 24>(x,    W1, b1, row, col, dinv, bufX, bufG, bufH, n, E, stream);
    run_layer< 24, 12>(bufH, W2, b2, row, col, dinv, bufX, bufG, bufH, n, E, stream);
    run_layer< 12,  6>(bufH, W3, b3, row, col, dinv, bufX, bufG, bufH, n, E, stream);
    run_layer<  6,  4>(bufH, W4, b4, row, col, dinv, bufX, bufG, bufH, n, E, stream);
    run_layer<  4,  2>(bufH, W5, b5, row, col, dinv, bufX, bufG, bufH, n, E, stream);

    float* out   = (float*)d_out;                   // [n, 4]
    float* hcopy = out + (size_t)n * 4;             // [n, 2]
    classifier_kernel<<<(n + THREADS - 1) / THREADS, THREADS, 0, stream>>>(bufH, Wc, bc, out, hcopy, n);
}